// WMOSA_37117107372441
// MI455X (gfx1250) — compile-verified
//
#include <hip/hip_runtime.h>
#include <hip/hip_bf16.h>

// ---------------------------------------------------------------------------
// WMOSA on MI455X (gfx1250): fused fp32 WMMA pipeline.
// Shapes: B=2, H=W=384, C=96, HEADS=6, HEAD_DIM=16, WIN=16 -> Bw=1152, L=256, k=64
// ---------------------------------------------------------------------------

#define BATCH   2
#define IMG     384
#define C_DIM   96
#define HEADS   6
#define HD      16
#define WIN     16
#define NWIN    24                  // 384/16
#define BW      (BATCH * NWIN * NWIN)   // 1152 windows
#define LTOK    256                 // tokens per window
#define KSEL    64                  // selected tokens per head
#define NPIX    (BATCH * IMG * IMG) // 294912

typedef __attribute__((ext_vector_type(2))) float v2f;
typedef __attribute__((ext_vector_type(8))) float v8f;

// fp32 WMMA 16x16x4 : D = A(16x4) * B(4x16) + C  (one tile per wave32)
static __device__ inline v8f wmma4(v2f a, v2f b, v8f c) {
    return __builtin_amdgcn_wmma_f32_16x16x4_f32(
        /*neg_a=*/false, a, /*neg_b=*/false, b,
        /*c_mod=*/(short)0, c, /*reuse_a=*/false, /*reuse_b=*/false);
}

// ---------------------------------------------------------------------------
// Kernel 1: router logits + sigmoid + include-first top-64 per head.
// One 256-thread block per window. Bitonic sort on (value, ~index) keys gives
// exact descending order with ties broken by lower index (matches lax.top_k).
// ---------------------------------------------------------------------------
__global__ __launch_bounds__(256) void k1_router(
    const float* __restrict__ x, const float* __restrict__ Wr,
    int* __restrict__ gidx, float* __restrict__ gval)
{
    __shared__ float sWr[C_DIM * HEADS];           // 96x6
    __shared__ float sv[LTOK];
    __shared__ unsigned long long key[LTOK];

    const int w = blockIdx.x;
    const int t = threadIdx.x;

    for (int i = t; i < C_DIM * HEADS; i += 256) sWr[i] = Wr[i];
    __syncthreads();

    const int b  = w / (NWIN * NWIN);
    const int rm = w % (NWIN * NWIN);
    const int wy = rm / NWIN, wx = rm % NWIN;
    const int hh = wy * WIN + (t >> 4);
    const int ww = wx * WIN + (t & 15);
    const float* xr = x + ((size_t)(b * IMG + hh) * IMG + ww) * C_DIM;

    float lg[HEADS];
    #pragma unroll
    for (int e = 0; e < HEADS; ++e) lg[e] = 0.f;
    for (int c = 0; c < C_DIM; ++c) {
        float xv = xr[c];
        #pragma unroll
        for (int e = 0; e < HEADS; ++e) lg[e] += xv * sWr[c * HEADS + e];
    }
    #pragma unroll
    for (int e = 0; e < HEADS; ++e) lg[e] = 1.0f / (1.0f + __expf(-lg[e]));

    for (int e = 0; e < HEADS; ++e) {
        __syncthreads();
        sv[t] = lg[e];
        // token 0 always first: force +INF key; sigmoid output > 0 so raw
        // fp32 bit pattern is order-preserving for the rest.
        unsigned int fb = (t == 0) ? 0x7F800000u : __float_as_uint(lg[e]);
        key[t] = ((unsigned long long)fb << 32) | (unsigned int)(0xFFFFu - t);
        __syncthreads();
        // bitonic sort, descending
        for (int ks = 2; ks <= LTOK; ks <<= 1) {
            for (int j = ks >> 1; j > 0; j >>= 1) {
                int ixj = t ^ j;
                if (ixj > t) {
                    unsigned long long a = key[t], bb = key[ixj];
                    bool descBlock = ((t & ks) == 0);
                    if (descBlock ? (a < bb) : (a > bb)) { key[t] = bb; key[ixj] = a; }
                }
                __syncthreads();
            }
        }
        if (t < KSEL) {
            int tok = 0xFFFF - (int)(key[t] & 0xFFFFull);
            gidx[((size_t)w * HEADS + e) * KSEL + t] = tok;
            gval[((size_t)w * HEADS + e) * KSEL + t] = sv[tok];
        }
    }
}

// ---------------------------------------------------------------------------
// Kernel 2: fused gather -> QKV -> attention -> Wo -> atomic scatter.
// grid (BW, HEADS); 128 threads = 4 wave32s; all GEMMs via v_wmma_f32_16x16x4.
// ---------------------------------------------------------------------------
__global__ __launch_bounds__(128) void k2_attn(
    const float* __restrict__ x,
    const float* __restrict__ Wqkv, const float* __restrict__ Wo,
    const int* __restrict__ gidx, const float* __restrict__ gval,
    float* __restrict__ scat)
{
    // phase-shared LDS pool:
    //  phase1: Xg[64][96] (6144) | Wqkv_e[96][48] (4608)
    //  phase2: S[64][64] (4096)  | O[64][16] (1024) | Wo_e[16][96] (1536)
    __shared__ float pool[11264];            // 44 KB
    __shared__ float sQKV[KSEL * 48];        // 12 KB
    __shared__ float sgate[KSEL];
    __shared__ int   sidx[KSEL];

    float* Xg  = pool;
    float* sWq = pool + 6144;
    float* S   = pool;
    float* O   = pool + 4096;
    float* sWo = pool + 5120;

    const int w    = blockIdx.x;
    const int e    = blockIdx.y;
    const int tid  = threadIdx.x;
    const int lane = tid & 31;
    const int wave = tid >> 5;               // 0..3
    const int hi   = lane >> 4;              // 0/1
    const int lo   = lane & 15;

    if (tid < KSEL) {
        sidx[tid]  = gidx[((size_t)w * HEADS + e) * KSEL + tid];
        sgate[tid] = gval[((size_t)w * HEADS + e) * KSEL + tid];
    }
    __syncthreads();

    const int b  = w / (NWIN * NWIN);
    const int rm = w % (NWIN * NWIN);
    const int wy = rm / NWIN, wx = rm % NWIN;

    // gather selected tokens: Xg[64][96]
    for (int i = tid; i < KSEL * C_DIM; i += 128) {
        int j = i / C_DIM, c = i % C_DIM;
        int t  = sidx[j];
        int hh = wy * WIN + (t >> 4);
        int ww = wx * WIN + (t & 15);
        Xg[i] = x[((size_t)(b * IMG + hh) * IMG + ww) * C_DIM + c];
    }
    for (int i = tid; i < C_DIM * 48; i += 128)
        sWq[i] = Wqkv[(size_t)e * C_DIM * 48 + i];
    __syncthreads();

    // QKV = Xg(64x96) @ Wqkv_e(96x48): 12 tiles, 3 per wave
    for (int tile = wave; tile < 12; tile += 4) {
        int mt = tile / 3, nt = tile % 3;
        v8f acc = {};
        for (int kk = 0; kk < C_DIM; kk += 4) {
            int ka = kk + 2 * hi;
            v2f a = { Xg[(mt * 16 + lo) * C_DIM + ka],
                      Xg[(mt * 16 + lo) * C_DIM + ka + 1] };
            v2f bf = { sWq[ka * 48 + nt * 16 + lo],
                       sWq[(ka + 1) * 48 + nt * 16 + lo] };
            acc = wmma4(a, bf, acc);
        }
        #pragma unroll
        for (int r = 0; r < 8; ++r)
            sQKV[(mt * 16 + r + 8 * hi) * 48 + nt * 16 + lo] = acc[r];
    }
    __syncthreads();

    // Wo_e load overlaps with the scores GEMM (disjoint pool regions)
    for (int i = tid; i < HD * C_DIM; i += 128)
        sWo[i] = Wo[(size_t)e * HD * C_DIM + i];

    // S = Q @ K^T * rsqrt(HD); Q = cols 0..15, K = cols 16..31 of sQKV
    for (int tile = wave; tile < 16; tile += 4) {
        int mt = tile / 4, nt = tile % 4;
        v8f acc = {};
        for (int kk = 0; kk < HD; kk += 4) {
            int ka = kk + 2 * hi;
            v2f a = { sQKV[(mt * 16 + lo) * 48 + ka],
                      sQKV[(mt * 16 + lo) * 48 + ka + 1] };
            v2f bf = { sQKV[(nt * 16 + lo) * 48 + 16 + ka],
                       sQKV[(nt * 16 + lo) * 48 + 16 + ka + 1] };
            acc = wmma4(a, bf, acc);
        }
        #pragma unroll
        for (int r = 0; r < 8; ++r)
            S[(mt * 16 + r + 8 * hi) * 64 + nt * 16 + lo] = acc[r] * 0.25f;
    }
    __syncthreads();

    // row softmax + router gate (gate folded into probabilities)
    if (tid < KSEL) {
        float m = -1e30f;
        for (int j = 0; j < 64; ++j) m = fmaxf(m, S[tid * 64 + j]);
        float s = 0.f;
        for (int j = 0; j < 64; ++j) {
            float ev = __expf(S[tid * 64 + j] - m);
            S[tid * 64 + j] = ev; s += ev;
        }
        float inv = sgate[tid] / s;
        for (int j = 0; j < 64; ++j) S[tid * 64 + j] *= inv;
    }
    __syncthreads();

    // O = P(64x64) @ V(64x16); V = cols 32..47 of sQKV; one tile per wave
    {
        int mt = wave;
        v8f acc = {};
        for (int kk = 0; kk < 64; kk += 4) {
            int ka = kk + 2 * hi;
            v2f a = { S[(mt * 16 + lo) * 64 + ka],
                      S[(mt * 16 + lo) * 64 + ka + 1] };
            v2f bf = { sQKV[ka * 48 + 32 + lo],
                       sQKV[(ka + 1) * 48 + 32 + lo] };
            acc = wmma4(a, bf, acc);
        }
        #pragma unroll
        for (int r = 0; r < 8; ++r)
            O[(mt * 16 + r + 8 * hi) * HD + lo] = acc[r];
    }
    __syncthreads();

    // pre = O(64x16) @ Wo_e(16x96); scatter-add to global (cross-head collisions)
    for (int tile = wave; tile < 24; tile += 4) {
        int mt = tile / 6, nt = tile % 6;
        v8f acc = {};
        for (int kk = 0; kk < HD; kk += 4) {
            int ka = kk + 2 * hi;
            v2f a = { O[(mt * 16 + lo) * HD + ka],
                      O[(mt * 16 + lo) * HD + ka + 1] };
            v2f bf = { sWo[ka * C_DIM + nt * 16 + lo],
                       sWo[(ka + 1) * C_DIM + nt * 16 + lo] };
            acc = wmma4(a, bf, acc);
        }
        #pragma unroll
        for (int r = 0; r < 8; ++r) {
            int m  = mt * 16 + r + 8 * hi;
            int t  = sidx[m];
            int hh = wy * WIN + (t >> 4);
            int ww = wx * WIN + (t & 15);
            atomicAdd(&scat[((size_t)(b * IMG + hh) * IMG + ww) * C_DIM + nt * 16 + lo],
                      acc[r]);
        }
    }
}

// ---------------------------------------------------------------------------
// Kernel 3: y = scat @ projW^T + projb. 64 rows per block, 8 waves, WMMA.
// projW (36 KB) stays L2-resident across all 4608 blocks.
// ---------------------------------------------------------------------------
__global__ __launch_bounds__(256) void k3_proj(
    const float* __restrict__ scat, const float* __restrict__ projW,
    const float* __restrict__ projb, float* __restrict__ y)
{
    __shared__ float A[64 * C_DIM];     // 24 KB
    __shared__ float Wt[C_DIM * C_DIM]; // 36 KB, Wt[c][j] = projW[j][c]
    __shared__ float bias[C_DIM];

    const int tid  = threadIdx.x;
    const size_t row0 = (size_t)blockIdx.x * 64;

    for (int i = tid; i < 64 * C_DIM; i += 256) A[i] = scat[row0 * C_DIM + i];
    for (int i = tid; i < C_DIM * C_DIM; i += 256) {
        int c = i / C_DIM, j = i % C_DIM;
        Wt[i] = projW[(size_t)j * C_DIM + c];
    }
    if (tid < C_DIM) bias[tid] = projb[tid];
    __syncthreads();

    const int lane = tid & 31, wave = tid >> 5;
    const int hi = lane >> 4, lo = lane & 15;

    for (int tile = wave; tile < 24; tile += 8) {   // 4 m-tiles x 6 n-tiles
        int mt = tile / 6, nt = tile % 6;
        v8f acc = {};
        for (int kk = 0; kk < C_DIM; kk += 4) {
            int ka = kk + 2 * hi;
            v2f a = { A[(mt * 16 + lo) * C_DIM + ka],
                      A[(mt * 16 + lo) * C_DIM + ka + 1] };
            v2f bf = { Wt[ka * C_DIM + nt * 16 + lo],
                       Wt[(ka + 1) * C_DIM + nt * 16 + lo] };
            acc = wmma4(a, bf, acc);
        }
        #pragma unroll
        for (int r = 0; r < 8; ++r) {
            int m = mt * 16 + r + 8 * hi;
            y[(row0 + m) * C_DIM + nt * 16 + lo] = acc[r] + bias[nt * 16 + lo];
        }
    }
}

// ---------------------------------------------------------------------------
extern "C" void kernel_launch(void* const* d_in, const int* in_sizes, int n_in,
                              void* d_out, int out_size, void* d_ws, size_t ws_size,
                              hipStream_t stream) {
    const float* x     = (const float*)d_in[0];   // [2,384,384,96]
    const float* Wr    = (const float*)d_in[1];   // [96,6]
    const float* Wqkv  = (const float*)d_in[2];   // [6,96,48]
    const float* Wo    = (const float*)d_in[3];   // [6,16,96]
    const float* projW = (const float*)d_in[4];   // [96,96]
    const float* projb = (const float*)d_in[5];   // [96]
    float* y = (float*)d_out;

    // workspace layout: scat [NPIX*C] f32, then gidx/gval [BW*HEADS*KSEL]
    const size_t scatBytes = (size_t)NPIX * C_DIM * sizeof(float);  // ~113 MB
    float* scat = (float*)d_ws;
    int*   gidx = (int*)((char*)d_ws + scatBytes);
    float* gval = (float*)((char*)d_ws + scatBytes +
                           (size_t)BW * HEADS * KSEL * sizeof(int));

    hipMemsetAsync(scat, 0, scatBytes, stream);   // zero scatter accumulator

    k1_router<<<BW, 256, 0, stream>>>(x, Wr, gidx, gval);
    k2_attn<<<dim3(BW, HEADS), 128, 0, stream>>>(x, Wqkv, Wo, gidx, gval, scat);
    k3_proj<<<NPIX / 64, 256, 0, stream>>>(scat, projW, projb, y);
}